// RoadMaterialsModel_37855841747607
// MI455X (gfx1250) — compile-verified
//
#include <hip/hip_runtime.h>
#include <math.h>

#define N_NODES 100000
#define N_EDGES 1600000
#define HID 128
#define EPS_BN 1e-3f
#define CHUNK 1024
#define NCHUNK 98   // ceil(100000/1024)

typedef float v2f __attribute__((ext_vector_type(2)));
typedef float v8f __attribute__((ext_vector_type(8)));

struct DP { const float *W, *b, *gamma, *beta, *mean, *var; };

// ---------------- utility ----------------
__global__ void k_zero(float* __restrict__ p, int n) {
    int i = blockIdx.x * blockDim.x + threadIdx.x;
    if (i < n) p[i] = 0.f;
}

// ---------------- edge stats: deg[n], S[n][0..7] = sum of edge_feat over edges with row==n --
__global__ void k_edge_stats(const float* __restrict__ ef, const int* __restrict__ eidx,
                             int* __restrict__ deg, float* __restrict__ S) {
    int e = blockIdx.x * blockDim.x + threadIdx.x;
    if (e >= N_EDGES) return;
    int r = eidx[e];                       // row
    atomicAdd(&deg[r], 1);
    const float* f = ef + (size_t)e * 8;
#pragma unroll
    for (int k = 0; k < 8; ++k) atomicAdd(&S[(size_t)r * 8 + k], f[k]);
}

// ---------------- scan stage 1: per-chunk sums ----------------
__global__ void k_chunk_sum(const int* __restrict__ deg, int* __restrict__ csum) {
    __shared__ int sm[256];
    int t = threadIdx.x;
    int base = blockIdx.x * CHUNK + t * 4;
    int s = 0;
#pragma unroll
    for (int j = 0; j < 4; ++j) { int idx = base + j; if (idx < N_NODES) s += deg[idx]; }
    sm[t] = s; __syncthreads();
    for (int off = 128; off > 0; off >>= 1) { if (t < off) sm[t] += sm[t + off]; __syncthreads(); }
    if (t == 0) csum[blockIdx.x] = sm[0];
}

// ---------------- scan stage 2: scan chunk sums (single block, 128 threads) ----------------
__global__ void k_scan_chunks(const int* __restrict__ csum, int* __restrict__ coff,
                              int* __restrict__ row_ptr, int nchunks) {
    __shared__ int sm[128];
    int t = threadIdx.x;
    int v = (t < nchunks) ? csum[t] : 0;
    sm[t] = v; __syncthreads();
    for (int off = 1; off < 128; off <<= 1) {
        int u = (t >= off) ? sm[t - off] : 0; __syncthreads();
        sm[t] += u; __syncthreads();
    }
    if (t < nchunks) coff[t] = sm[t] - v;      // exclusive
    if (t == 127) row_ptr[N_NODES] = sm[127];  // == E
}

// ---------------- scan stage 3: per-chunk exclusive scan -> row_ptr, cursor ----------------
__global__ void k_chunk_scan(const int* __restrict__ deg, const int* __restrict__ coff,
                             int* __restrict__ row_ptr, int* __restrict__ cursor) {
    __shared__ int sm[256];
    int t = threadIdx.x;
    int base = blockIdx.x * CHUNK + t * 4;
    int d[4]; int s = 0;
#pragma unroll
    for (int j = 0; j < 4; ++j) { int idx = base + j; d[j] = (idx < N_NODES) ? deg[idx] : 0; s += d[j]; }
    sm[t] = s; __syncthreads();
    for (int off = 1; off < 256; off <<= 1) {
        int u = (t >= off) ? sm[t - off] : 0; __syncthreads();
        sm[t] += u; __syncthreads();
    }
    int run = coff[blockIdx.x] + sm[t] - s;
#pragma unroll
    for (int j = 0; j < 4; ++j) {
        int idx = base + j;
        if (idx < N_NODES) { row_ptr[idx] = run; cursor[idx] = run; run += d[j]; }
    }
}

// ---------------- CSR fill ----------------
__global__ void k_fill_csr(const int* __restrict__ eidx, int* __restrict__ cursor,
                           int* __restrict__ cols) {
    int e = blockIdx.x * blockDim.x + threadIdx.x;
    if (e >= N_EDGES) return;
    int r = eidx[e];
    int c = eidx[N_EDGES + e];
    int pos = atomicAdd(&cursor[r], 1);
    cols[pos] = c;
}

// ---------------- x0 = node_feat @ node_W + node_b  (WMMA f32 16x16x4, K=16) ----------------
__global__ void k_node_xform(const float* __restrict__ nf, const float* __restrict__ W,
                             const float* __restrict__ b, float* __restrict__ x) {
    int wave = threadIdx.x >> 5, lane = threadIdx.x & 31;
    int m0 = blockIdx.x * 16, n0 = wave * 16;
    int m = lane & 15, hl = lane >> 4;
    v8f acc = {};
#pragma unroll
    for (int kb = 0; kb < 16; kb += 4) {
        int ka = kb + 2 * hl;
        v2f a, bb;
        a.x  = nf[(size_t)(m0 + m) * 16 + ka];
        a.y  = nf[(size_t)(m0 + m) * 16 + ka + 1];
        bb.x = W[(size_t)ka * HID + n0 + m];
        bb.y = W[(size_t)(ka + 1) * HID + n0 + m];
        acc = __builtin_amdgcn_wmma_f32_16x16x4_f32(false, a, false, bb, (short)0, acc, false, false);
    }
    int col = n0 + m;
    float bias = b[col];
#pragma unroll
    for (int r = 0; r < 8; ++r) {
        int row = m0 + r + 8 * hl;
        x[(size_t)row * HID + col] = acc[r] + bias;
    }
}

// -------- per-layer aggregation: agg[n] = deg*edge_b + S[n]@edge_W + sum_{e in n} x[col[e]] --
__global__ void k_aggregate(const float* __restrict__ x, const float* __restrict__ S,
                            const int* __restrict__ row_ptr, const int* __restrict__ cols,
                            const float* __restrict__ eW, const float* __restrict__ eb,
                            float* __restrict__ agg) {
    int n = blockIdx.x * 8 + (threadIdx.x >> 5);
    int lane = threadIdx.x & 31;
    if (n >= N_NODES) return;
    int start = row_ptr[n], end = row_ptr[n + 1];
    float dg = (float)(end - start);
    int c4 = lane * 4;
    float a0 = dg * eb[c4], a1 = dg * eb[c4 + 1], a2 = dg * eb[c4 + 2], a3 = dg * eb[c4 + 3];
#pragma unroll
    for (int k = 0; k < 8; ++k) {
        float sk = S[(size_t)n * 8 + k];
        const float* w = eW + (size_t)k * HID + c4;
        a0 += sk * w[0]; a1 += sk * w[1]; a2 += sk * w[2]; a3 += sk * w[3];
    }
    for (int e = start; e < end; ++e) {
        int c = cols[e];
        if (e + 1 < end) {
            int cn = cols[e + 1];
            __builtin_prefetch(x + (size_t)cn * HID + c4, 0, 0);   // global_prefetch
        }
        const float4 v = *(const float4*)(x + (size_t)c * HID + c4);
        a0 += v.x; a1 += v.y; a2 += v.z; a3 += v.w;
    }
    float4 out; out.x = a0; out.y = a1; out.z = a2; out.w = a3;
    *(float4*)(agg + (size_t)n * HID + c4) = out;
}

// -------- x += BN(relu(agg @ W + b))  (WMMA f32 16x16x4, K=128) ----------------
__global__ void k_conv_gemm(const float* __restrict__ agg, const float* __restrict__ W,
                            const float* __restrict__ b, const float* __restrict__ gamma,
                            const float* __restrict__ beta, const float* __restrict__ mean,
                            const float* __restrict__ var, float* __restrict__ x) {
    int wave = threadIdx.x >> 5, lane = threadIdx.x & 31;
    int m0 = blockIdx.x * 16, n0 = wave * 16;
    int m = lane & 15, hl = lane >> 4;
    v8f acc = {};
    const float* arow = agg + (size_t)(m0 + m) * HID + 2 * hl;
#pragma unroll 4
    for (int kb = 0; kb < HID; kb += 4) {
        v2f a = *(const v2f*)(arow + kb);          // K = kb+2hl, kb+2hl+1
        int ka = kb + 2 * hl;
        v2f bb;
        bb.x = W[(size_t)ka * HID + n0 + m];
        bb.y = W[(size_t)(ka + 1) * HID + n0 + m];
        acc = __builtin_amdgcn_wmma_f32_16x16x4_f32(false, a, false, bb, (short)0, acc, false, false);
    }
    int col = n0 + m;
    float bias = b[col], g = gamma[col], bt = beta[col], mu = mean[col];
    float rs = rsqrtf(var[col] + EPS_BN);
#pragma unroll
    for (int r = 0; r < 8; ++r) {
        int row = m0 + r + 8 * hl;
        float v = acc[r] + bias;
        v = v > 0.f ? v : 0.f;
        v = (v - mu) * rs * g + bt;
        x[(size_t)row * HID + col] += v;
    }
}

// ---------------- column sums of x ----------------
__global__ void k_colsum(const float* __restrict__ x, float* __restrict__ colsum) {
    int t = threadIdx.x;   // 128
    float acc = 0.f;
    for (int n = blockIdx.x; n < N_NODES; n += gridDim.x) acc += x[(size_t)n * HID + t];
    atomicAdd(&colsum[t], acc);
}

// ---------------- dense head (single block, 128 threads) ----------------
__global__ void k_head(const float* __restrict__ colsum, DP d0, DP d1, DP d2,
                       const float* __restrict__ fW, const float* __restrict__ fb,
                       float* __restrict__ out) {
    __shared__ float h[HID];
    int t = threadIdx.x;
    h[t] = colsum[t] / (float)N_NODES;
    __syncthreads();
    DP dps[3] = {d0, d1, d2};
    for (int l = 0; l < 3; ++l) {
        DP p = dps[l];
        float v = p.b[t];
        for (int k = 0; k < HID; ++k) v += h[k] * p.W[k * HID + t];
        v = v > 0.f ? v : 0.f;
        v = (v - p.mean[t]) * rsqrtf(p.var[t] + EPS_BN) * p.gamma[t] + p.beta[t];
        __syncthreads();
        h[t] = v;
        __syncthreads();
    }
    if (t < 3) {
        float v = fb[t];
        for (int k = 0; k < HID; ++k) v += h[k] * fW[k * 3 + t];
        out[t] = v;
    }
}

extern "C" void kernel_launch(void* const* d_in, const int* in_sizes, int n_in,
                              void* d_out, int out_size, void* d_ws, size_t ws_size,
                              hipStream_t stream) {
    (void)n_in; (void)out_size; (void)ws_size;
    const float *node_feat, *edge_feat, *node_W, *node_b, *edge_W, *edge_b, *final_W, *final_b;
    const int* edge_index;
    DP conv[4], dense[3];

    auto F = [&](int i) { return (const float*)d_in[i]; };

    if (in_sizes[0] == N_NODES * 16) {
        // insertion-order flattening
        node_feat = F(0); edge_feat = F(1); edge_index = (const int*)d_in[2];
        node_W = F(3); node_b = F(4); edge_W = F(5); edge_b = F(6);
        for (int i = 0; i < 4; ++i) {
            int base = 7 + 6 * i;  // W, b, gamma, beta, mean, var
            conv[i] = {F(base), F(base + 1), F(base + 2), F(base + 3), F(base + 4), F(base + 5)};
        }
        for (int i = 0; i < 3; ++i) {
            int base = 31 + 6 * i;
            dense[i] = {F(base), F(base + 1), F(base + 2), F(base + 3), F(base + 4), F(base + 5)};
        }
        final_W = F(49); final_b = F(50);
    } else {
        // jax-pytree (alphabetically sorted) flattening
        edge_feat = F(0); edge_index = (const int*)d_in[1]; node_feat = F(2);
        for (int i = 0; i < 4; ++i) {
            int base = 3 + 6 * i;  // W, b, beta, gamma, mean, var
            conv[i] = {F(base), F(base + 1), F(base + 3), F(base + 2), F(base + 4), F(base + 5)};
        }
        for (int i = 0; i < 3; ++i) {
            int base = 27 + 6 * i;
            dense[i] = {F(base), F(base + 1), F(base + 3), F(base + 2), F(base + 4), F(base + 5)};
        }
        edge_W = F(45); edge_b = F(46); final_W = F(47); final_b = F(48);
        node_W = F(49); node_b = F(50);
    }

    // ---- workspace carve ----
    char* ws = (char*)d_ws;
    float* x   = (float*)ws;            ws += (size_t)N_NODES * HID * 4;
    float* agg = (float*)ws;            ws += (size_t)N_NODES * HID * 4;
    // zero-region: deg | S | colsum (contiguous)
    int*   deg    = (int*)ws;           ws += (size_t)N_NODES * 4;
    float* S      = (float*)ws;         ws += (size_t)N_NODES * 8 * 4;
    float* colsum = (float*)ws;         ws += 128 * 4;
    int* row_ptr = (int*)ws;            ws += ((size_t)N_NODES + 4) * 4;   // padded
    int* cursor  = (int*)ws;            ws += (size_t)N_NODES * 4;
    int* cols    = (int*)ws;            ws += (size_t)N_EDGES * 4;
    int* csum    = (int*)ws;            ws += 128 * 4;
    int* coff    = (int*)ws;            ws += 128 * 4;

    const int zcount = N_NODES + N_NODES * 8 + 128;
    k_zero<<<(zcount + 255) / 256, 256, 0, stream>>>((float*)deg, zcount);

    k_edge_stats<<<(N_EDGES + 255) / 256, 256, 0, stream>>>(edge_feat, edge_index, deg, S);
    k_chunk_sum<<<NCHUNK, 256, 0, stream>>>(deg, csum);
    k_scan_chunks<<<1, 128, 0, stream>>>(csum, coff, row_ptr, NCHUNK);
    k_chunk_scan<<<NCHUNK, 256, 0, stream>>>(deg, coff, row_ptr, cursor);
    k_fill_csr<<<(N_EDGES + 255) / 256, 256, 0, stream>>>(edge_index, cursor, cols);

    k_node_xform<<<N_NODES / 16, 256, 0, stream>>>(node_feat, node_W, node_b, x);

    for (int l = 0; l < 4; ++l) {
        k_aggregate<<<N_NODES / 8, 256, 0, stream>>>(x, S, row_ptr, cols, edge_W, edge_b, agg);
        k_conv_gemm<<<N_NODES / 16, 256, 0, stream>>>(agg, conv[l].W, conv[l].b, conv[l].gamma,
                                                      conv[l].beta, conv[l].mean, conv[l].var, x);
    }

    k_colsum<<<256, 128, 0, stream>>>(x, colsum);
    k_head<<<1, 128, 0, stream>>>(colsum, dense[0], dense[1], dense[2], final_W, final_b,
                                  (float*)d_out);
}